// GNN_4234837753916
// MI455X (gfx1250) — compile-verified
//
#include <hip/hip_runtime.h>
#include <hip/hip_bf16.h>
#include <math.h>

#define NEG_SLOPE 0.02f

typedef __attribute__((ext_vector_type(2))) float v2f;
typedef __attribute__((ext_vector_type(8))) float v8f;

// ---------------- degree / norm ----------------
__global__ void k_deg_init(float* deg, int n) {
    int i = blockIdx.x * blockDim.x + threadIdx.x;
    if (i < n) deg[i] = 1.0f;  // self-loop
}

__global__ void k_deg_edges(const int* __restrict__ dst, float* deg, int E) {
    int e = blockIdx.x * blockDim.x + threadIdx.x;
    if (e < E) atomicAdd(&deg[dst[e]], 1.0f);
}

__global__ void k_deg_rsqrt(float* deg, int n) {
    int i = blockIdx.x * blockDim.x + threadIdx.x;
    if (i < n) deg[i] = rsqrtf(deg[i]);   // deg >= 1 always
}

// ---------------- dense GEMM via f32 WMMA ----------------
// H[n x 16] = X[n x kdim] @ W[kdim x 16]; one wave32 per 16-row tile.
// n must be a multiple of 16 (true: 200000). kdim in {128, 16}, multiple of 4.
__global__ void k_gemm16_wmma(const float* __restrict__ X,
                              const float* __restrict__ W,
                              float* __restrict__ H, int n, int kdim) {
    __shared__ float sW[128 * 16];
    int t = threadIdx.x;
    for (int i = t; i < kdim * 16; i += blockDim.x) sW[i] = W[i];
    __syncthreads();

    int wave = t >> 5;
    int lane = t & 31;
    int tile = blockIdx.x * (blockDim.x >> 5) + wave;   // wave-uniform
    int row0 = tile * 16;
    if (row0 >= n) return;                              // whole wave exits together

    int half = lane >> 4;       // 0: K pair {0,1}; 1: K pair {2,3}
    int lr   = lane & 15;
    long rowbase = (long)(row0 + lr) * kdim;

    v8f c = {};
    for (int k0 = 0; k0 < kdim; k0 += 4) {
        int ka = k0 + 2 * half;
        // A 16x4: lane(lr) = row M, (VGPR0,VGPR1) = K = ka, ka+1
        v2f a = *(const v2f*)(X + rowbase + ka);
        // B 4x16: lane(lr) = col N, (VGPR0,VGPR1) = K = ka, ka+1
        v2f b;
        b.x = sW[(ka + 0) * 16 + lr];
        b.y = sW[(ka + 1) * 16 + lr];
        c = __builtin_amdgcn_wmma_f32_16x16x4_f32(
                /*neg_a=*/false, a, /*neg_b=*/false, b,
                /*c_mod=*/(short)0, c, /*reuse_a=*/false, /*reuse_b=*/false);
    }
    // C/D layout: VGPR v -> row M = v + 8*half, col = lr
#pragma unroll
    for (int v = 0; v < 8; ++v)
        H[(long)(row0 + v + 8 * half) * 16 + lr] = c[v];
}

// ---------------- self-loop term: agg = dinv^2 * h ----------------
__global__ void k_selfloop(const float* __restrict__ H, const float* __restrict__ dinv,
                           float* __restrict__ agg, int n16) {
    int i = blockIdx.x * blockDim.x + threadIdx.x;
    if (i >= n16) return;
    float di = dinv[i >> 4];
    agg[i] = di * di * H[i];
}

// ---------------- edge-parallel scatter: agg[d] += dinv[s]*dinv[d]*h[s] ----------------
__global__ void k_edge_scatter(const float* __restrict__ H,
                               const int* __restrict__ src, const int* __restrict__ dst,
                               const float* __restrict__ dinv,
                               float* __restrict__ agg, int E) {
    int e = blockIdx.x * blockDim.x + threadIdx.x;
    if (e >= E) return;
    int s = src[e], d = dst[e];
    float w = dinv[s] * dinv[d];
    const float4* hs = (const float4*)(H + (long)s * 16);
    float* out = agg + (long)d * 16;
#pragma unroll
    for (int q = 0; q < 4; ++q) {
        float4 hv = hs[q];
        atomicAdd(out + 4 * q + 0, w * hv.x);
        atomicAdd(out + 4 * q + 1, w * hv.y);
        atomicAdd(out + 4 * q + 2, w * hv.z);
        atomicAdd(out + 4 * q + 3, w * hv.w);
    }
}

// ---------------- bias + optional ReLU (in place) ----------------
__global__ void k_bias_act(float* __restrict__ agg, const float* __restrict__ b,
                           int n16, int do_relu) {
    int i = blockIdx.x * blockDim.x + threadIdx.x;
    if (i >= n16) return;
    float v = agg[i] + b[i & 15];
    if (do_relu) v = v > 0.0f ? v : 0.0f;
    agg[i] = v;
}

// ---------------- fused MLP (16->64->16->2) + 2-class log-softmax ----------------
__global__ void k_mlp(const float* __restrict__ H,
                      const float* __restrict__ M1, const float* __restrict__ mb1,
                      const float* __restrict__ M2, const float* __restrict__ mb2,
                      const float* __restrict__ M3, const float* __restrict__ mb3,
                      float* __restrict__ out, int n) {
    __shared__ float sM1[16 * 64], sB1[64], sM2[64 * 16], sB2[16], sM3[16 * 2], sB3[2];
    int t = threadIdx.x;
    for (int i = t; i < 16 * 64; i += blockDim.x) sM1[i] = M1[i];
    for (int i = t; i < 64;      i += blockDim.x) sB1[i] = mb1[i];
    for (int i = t; i < 64 * 16; i += blockDim.x) sM2[i] = M2[i];
    for (int i = t; i < 16;      i += blockDim.x) sB2[i] = mb2[i];
    for (int i = t; i < 32;      i += blockDim.x) sM3[i] = M3[i];
    for (int i = t; i < 2;       i += blockDim.x) sB3[i] = mb3[i];
    __syncthreads();

    int i = blockIdx.x * blockDim.x + t;
    if (i >= n) return;

    float h[16];
#pragma unroll
    for (int k = 0; k < 16; ++k) h[k] = H[(long)i * 16 + k];

    float a[64];
#pragma unroll 4
    for (int j = 0; j < 64; ++j) {
        float acc = sB1[j];
#pragma unroll
        for (int k = 0; k < 16; ++k) acc += h[k] * sM1[k * 64 + j];
        a[j] = acc > 0.0f ? acc : NEG_SLOPE * acc;
    }
    float c[16];
#pragma unroll 4
    for (int j = 0; j < 16; ++j) {
        float acc = sB2[j];
#pragma unroll
        for (int k = 0; k < 64; ++k) acc += a[k] * sM2[k * 16 + j];
        c[j] = acc > 0.0f ? acc : NEG_SLOPE * acc;
    }
    float l0 = sB3[0], l1 = sB3[1];
#pragma unroll
    for (int k = 0; k < 16; ++k) {
        l0 += c[k] * sM3[k * 2 + 0];
        l1 += c[k] * sM3[k * 2 + 1];
    }
    float m   = fmaxf(l0, l1);
    float lse = m + logf(expf(l0 - m) + expf(l1 - m));
    out[(long)i * 2 + 0] = l0 - lse;
    out[(long)i * 2 + 1] = l1 - lse;
}

extern "C" void kernel_launch(void* const* d_in, const int* in_sizes, int n_in,
                              void* d_out, int out_size, void* d_ws, size_t ws_size,
                              hipStream_t stream) {
    const float* x   = (const float*)d_in[0];
    const int*  data = (const int*)  d_in[1];
    const float* W1  = (const float*)d_in[2];
    const float* b1  = (const float*)d_in[3];
    const float* W2  = (const float*)d_in[4];
    const float* b2  = (const float*)d_in[5];
    const float* W3  = (const float*)d_in[6];
    const float* b3  = (const float*)d_in[7];
    const float* M1  = (const float*)d_in[8];
    const float* mb1 = (const float*)d_in[9];
    const float* M2  = (const float*)d_in[10];
    const float* mb2 = (const float*)d_in[11];
    const float* M3  = (const float*)d_in[12];
    const float* mb3 = (const float*)d_in[13];
    float* out = (float*)d_out;

    const int n   = in_sizes[0] / 128;   // 200000
    const int E   = in_sizes[1] / 2;     // 3200000
    const int n16 = n * 16;
    const int* src = data;
    const int* dst = data + E;

    float* wsf  = (float*)d_ws;
    float* dinv = wsf;               // n
    float* h    = wsf + n;           // n*16
    float* agg  = wsf + n + n16;     // n*16

    const int B = 256;
    dim3 blk(B);
    dim3 gN((n + B - 1) / B);
    dim3 gE((E + B - 1) / B);
    dim3 gN16((n16 + B - 1) / B);
    int tiles = n / 16;
    dim3 gGemm((tiles + (B / 32) - 1) / (B / 32));

    // degrees -> dinv
    k_deg_init<<<gN, blk, 0, stream>>>(dinv, n);
    k_deg_edges<<<gE, blk, 0, stream>>>(dst, dinv, E);
    k_deg_rsqrt<<<gN, blk, 0, stream>>>(dinv, n);

    // ---- layer 1: x(128) @ W1 -> aggregate -> +b1, relu ----
    k_gemm16_wmma<<<gGemm, blk, 0, stream>>>(x, W1, h, n, 128);
    k_selfloop<<<gN16, blk, 0, stream>>>(h, dinv, agg, n16);
    k_edge_scatter<<<gE, blk, 0, stream>>>(h, src, dst, dinv, agg, E);
    k_bias_act<<<gN16, blk, 0, stream>>>(agg, b1, n16, 1);

    // ---- layer 2 ----
    k_gemm16_wmma<<<gGemm, blk, 0, stream>>>(agg, W2, h, n, 16);
    k_selfloop<<<gN16, blk, 0, stream>>>(h, dinv, agg, n16);
    k_edge_scatter<<<gE, blk, 0, stream>>>(h, src, dst, dinv, agg, E);
    k_bias_act<<<gN16, blk, 0, stream>>>(agg, b2, n16, 1);

    // ---- layer 3 (no relu) ----
    k_gemm16_wmma<<<gGemm, blk, 0, stream>>>(agg, W3, h, n, 16);
    k_selfloop<<<gN16, blk, 0, stream>>>(h, dinv, agg, n16);
    k_edge_scatter<<<gE, blk, 0, stream>>>(h, src, dst, dinv, agg, E);
    k_bias_act<<<gN16, blk, 0, stream>>>(agg, b3, n16, 0);

    // ---- MLP + log_softmax ----
    k_mlp<<<gN, blk, 0, stream>>>(agg, M1, mb1, M2, mb2, M3, mb3, out, n);
}